// BalancedFrequencyAttention_8658654069233
// MI455X (gfx1250) — compile-verified
//
#include <hip/hip_runtime.h>
#include <math.h>

typedef __attribute__((ext_vector_type(2))) float v2f;
typedef __attribute__((ext_vector_type(8))) float v8f;

#define Hd   200
#define Wd   480
#define Cd   128
#define Bd   8
#define HWd  (Hd * Wd)     // 96000 elements per (b,c) slice
#define HW4  (HWd / 4)     // 24000 float4 per slice

// -------------------------------------------------------------------------
// Kernel 1: gap[b*C+c] = sum_{h,n} x[b,c,h,n] * G[class(h), n]
// G has only 4 distinct rows (closed-form cosines), built once per block
// in LDS with double-precision trig (1920 values, negligible cost).
// -------------------------------------------------------------------------
__global__ __launch_bounds__(256) void gap_kernel(const float* __restrict__ x,
                                                  float* __restrict__ gap) {
    __shared__ float g[4 * Wd];
    __shared__ float red[256];
    const int tid = threadIdx.x;

    const double PI   = 3.14159265358979323846;
    const double c0   = 1.0 / (200.0 * sqrt(240.0));     // 1/(H * sqrt(240)) mean factor
    const double base = 0.6 * c0 * sqrt(1.0 / 480.0);    // low path: 0.6 * D[0,n]
    const double scl  = 0.4 * c0 * sqrt(2.0 / 480.0);    // high path DCT row scale

    for (int i = tid; i < 4 * Wd; i += 256) {
        int cls = i / Wd, n = i % Wd;
        double ang = PI * (2.0 * n + 1.0) / 960.0;
        double v = base;
        if      (cls == 1) v += scl * (cos(ang *  80.0) + cos(ang * 320.0));
        else if (cls == 2) v += scl * (cos(ang * 160.0) + cos(ang * 400.0));
        else if (cls == 3) v += scl *  cos(ang * 240.0);
        g[i] = (float)v;
    }
    __syncthreads();

    const float4* xp = (const float4*)(x + (size_t)blockIdx.x * HWd);
    float acc = 0.f;
    #pragma unroll 4
    for (int i = tid; i < HW4; i += 256) {
        float4 v = xp[i];
        int h = i / (Wd / 4);
        int n = (i % (Wd / 4)) * 4;
        int cls = (h < 80) ? 0 : (1 + (h - 80) % 3);
        const float* gw = &g[cls * Wd + n];
        acc += v.x * gw[0] + v.y * gw[1] + v.z * gw[2] + v.w * gw[3];
    }
    red[tid] = acc;
    __syncthreads();
    for (int s = 128; s > 0; s >>= 1) {
        if (tid < s) red[tid] += red[tid + s];
        __syncthreads();
    }
    if (tid == 0) gap[blockIdx.x] = red[0];
}

// -------------------------------------------------------------------------
// Kernel 2: att = sigmoid(gap @ w1^T @ w2^T), one wave32, fp32 WMMA.
//   hidden[16x32] = gap[16x128] @ w1^T   (K=128, 2 N-tiles)
//   att   [16x128]= hidden      @ w2^T   (K=32,  8 N-tiles)
// Fragment layouts per ISA 7.12.2 (16x4 f32 A: lanes 0-15 hold K0/K1,
// lanes 16-31 hold K2/K3; C/D: VGPR v = row v (lanes 0-15) / row v+8).
// -------------------------------------------------------------------------
__global__ __launch_bounds__(32) void att_kernel(const float* __restrict__ gap,
                                                 const float* __restrict__ w1,
                                                 const float* __restrict__ w2,
                                                 float* __restrict__ att) {
    __shared__ float gapL[16 * Cd];   // rows 8..15 zero-padded
    __shared__ float w1L[32 * Cd];    // w1[j, c] row-major
    __shared__ float w2L[Cd * 32];    // w2[c, j] row-major
    __shared__ float hidL[16 * 32];

    const int lane = threadIdx.x;
    for (int i = lane; i < 16 * Cd; i += 32) gapL[i] = (i < Bd * Cd) ? gap[i] : 0.f;
    for (int i = lane; i < 32 * Cd; i += 32) w1L[i] = w1[i];
    for (int i = lane; i < Cd * 32; i += 32) w2L[i] = w2[i];
    __syncthreads();

    const int hi = lane >> 4;   // 0: K pair {0,1}, 1: K pair {2,3}
    const int l  = lane & 15;

    // GEMM1: B[k][n] = w1[n][k]
    for (int nt = 0; nt < 2; ++nt) {
        v8f acc = {0.f, 0.f, 0.f, 0.f, 0.f, 0.f, 0.f, 0.f};
        for (int k0 = 0; k0 < Cd; k0 += 4) {
            int ka = k0 + 2 * hi;
            v2f a, b;
            a.x = gapL[l * Cd + ka];
            a.y = gapL[l * Cd + ka + 1];
            b.x = w1L[(nt * 16 + l) * Cd + ka];
            b.y = w1L[(nt * 16 + l) * Cd + ka + 1];
            acc = __builtin_amdgcn_wmma_f32_16x16x4_f32(false, a, false, b,
                                                        (short)0, acc, false, false);
        }
        #pragma unroll
        for (int v = 0; v < 8; ++v)
            hidL[(hi * 8 + v) * 32 + nt * 16 + l] = acc[v];
    }
    __syncthreads();

    // GEMM2: B[k][n] = w2[n][k]; sigmoid; rows 0..7 are the real batch
    for (int nt = 0; nt < 8; ++nt) {
        v8f acc = {0.f, 0.f, 0.f, 0.f, 0.f, 0.f, 0.f, 0.f};
        for (int k0 = 0; k0 < 32; k0 += 4) {
            int ka = k0 + 2 * hi;
            v2f a, b;
            a.x = hidL[l * 32 + ka];
            a.y = hidL[l * 32 + ka + 1];
            b.x = w2L[(nt * 16 + l) * 32 + ka];
            b.y = w2L[(nt * 16 + l) * 32 + ka + 1];
            acc = __builtin_amdgcn_wmma_f32_16x16x4_f32(false, a, false, b,
                                                        (short)0, acc, false, false);
        }
        if (hi == 0) {
            #pragma unroll
            for (int v = 0; v < 8; ++v) {
                float t = acc[v];
                att[v * Cd + nt * 16 + l] = 1.f / (1.f + __expf(-t));
            }
        }
    }
}

// -------------------------------------------------------------------------
// Kernel 3: out = x * att[b*C+c], pure streaming, float4 per thread.
// -------------------------------------------------------------------------
__global__ __launch_bounds__(256) void scale_kernel(const float* __restrict__ x,
                                                    const float* __restrict__ att,
                                                    float* __restrict__ out) {
    unsigned i = blockIdx.x * 256u + threadIdx.x;   // one float4 per thread
    const float4* x4 = (const float4*)x;
    float4* o4 = (float4*)out;
    float a = att[i / HW4];
    float4 v = x4[i];
    v.x *= a; v.y *= a; v.z *= a; v.w *= a;
    o4[i] = v;
}

extern "C" void kernel_launch(void* const* d_in, const int* in_sizes, int n_in,
                              void* d_out, int out_size, void* d_ws, size_t ws_size,
                              hipStream_t stream) {
    (void)in_sizes; (void)n_in; (void)out_size; (void)ws_size;
    const float* x  = (const float*)d_in[0];
    const float* w1 = (const float*)d_in[1];
    const float* w2 = (const float*)d_in[2];
    float* out = (float*)d_out;
    float* gap = (float*)d_ws;          // 1024 floats
    float* att = gap + 1024;            // 1024 floats

    gap_kernel<<<Bd * Cd, 256, 0, stream>>>(x, gap);
    att_kernel<<<1, 32, 0, stream>>>(gap, w1, w2, att);
    const unsigned n4 = (unsigned)Bd * Cd * HW4;   // 24,576,000 float4
    scale_kernel<<<n4 / 256, 256, 0, stream>>>(x, att, out);
}